// QuantumBranch_49675591745616
// MI455X (gfx1250) — compile-verified
//
#include <hip/hip_runtime.h>
#include <math.h>

// MI455X / gfx1250, wave32. WMMA: V_WMMA_F32_16X16X4_F32 (fp32-exact K=4 GEMM).
// Streaming load/stores use non-temporal hints (CDNA5 TH=NT).

typedef __attribute__((ext_vector_type(2))) float v2f;
typedef __attribute__((ext_vector_type(4))) float v4f;
typedef __attribute__((ext_vector_type(8))) float v8f;

#define COEFF_OFF 32   // cons[0..19]=LN constants, cons[32..355]=Pauli coeffs (4x81)

// ---------------- fast math (hardware TRANS ops) ----------------
__device__ __forceinline__ float f_sin(float a) {
#if __has_builtin(__builtin_amdgcn_sinf)
  return __builtin_amdgcn_sinf(a * 0.15915494309189535f);  // v_sin: input in revolutions
#else
  return sinf(a);
#endif
}
__device__ __forceinline__ float f_cos(float a) {
#if __has_builtin(__builtin_amdgcn_cosf)
  return __builtin_amdgcn_cosf(a * 0.15915494309189535f);
#else
  return cosf(a);
#endif
}
__device__ __forceinline__ float f_exp(float a) {
#if __has_builtin(__builtin_amdgcn_exp2f)
  return __builtin_amdgcn_exp2f(a * 1.4426950408889634f);  // v_exp = 2^x
#else
  return expf(a);
#endif
}
__device__ __forceinline__ float f_rcp(float a) {
#if __has_builtin(__builtin_amdgcn_rcpf)
  return __builtin_amdgcn_rcpf(a);
#else
  return 1.f / a;
#endif
}
__device__ __forceinline__ float f_rsq(float a) {
#if __has_builtin(__builtin_amdgcn_rsqf)
  return __builtin_amdgcn_rsqf(a);
#else
  return rsqrtf(a);
#endif
}
__device__ __forceinline__ float f_tanh(float v) {
  float xc = fminf(fmaxf(v, -15.f), 15.f);
  float e2 = f_exp(2.f * xc);
  return (e2 - 1.f) * f_rcp(e2 + 1.f);
}

__device__ __forceinline__ float bperm(int srcLane, float v) {
  return __int_as_float(__builtin_amdgcn_ds_bpermute(srcLane << 2, __float_as_int(v)));
}

// Reverse-order gate list for Heisenberg conjugation M <- G^H M G.
// Forward circuit: ROT(0,0..3); CNOT(0,1)(1,2)(2,3)(3,0); ROT(1,0..3); CNOT(0,2)(1,3)(2,0)(3,1)
__device__ __constant__ int g_type[16] = {0,0,0,0, 1,1,1,1, 0,0,0,0, 1,1,1,1};
__device__ __constant__ int g_a[16]    = {3,2,1,0, 1,1,1,1, 3,2,1,0, 0,0,0,0};
__device__ __constant__ int g_b[16]    = {1,0,3,2, 3,2,1,0, 0,3,2,1, 3,2,1,0};

// ---------------- prep kernel (1 block, 64 threads) ----------------
// Produces: cons[0..19]  LayerNorm constants (sum W, W^T b, upper-tri W^T W, sum b, sum b^2)
//           cons[32+w*81+s]  Pauli coefficients: z_w = sum_s c * prod_u t_u[s_u],
//                            t_u = (1, -sin x_u, cos x_u), s_u in {I,Y,Z}
__global__ void __launch_bounds__(64)
qprep(const float* __restrict__ W, const float* __restrict__ b,
      const float* __restrict__ wts, float* __restrict__ cons) {
  __shared__ float Mr[4][16][16];
  __shared__ float Mi[4][16][16];
  __shared__ float red[20][64];
  const int t = threadIdx.x;

  // ---- LayerNorm constants ----
  {
    const int n = t;  // 64 = PD
    float w0 = W[n*4+0], w1 = W[n*4+1], w2 = W[n*4+2], w3 = W[n*4+3], bn = b[n];
    float v[20] = { w0, w1, w2, w3,
                    bn*w0, bn*w1, bn*w2, bn*w3,
                    w0*w0, w0*w1, w0*w2, w0*w3, w1*w1, w1*w2, w1*w3, w2*w2, w2*w3, w3*w3,
                    bn, bn*bn };
    #pragma unroll
    for (int k = 0; k < 20; ++k) red[k][n] = v[k];
    __syncthreads();
    if (t < 20) {
      float s = 0.f;
      for (int k = 0; k < 64; ++k) s += red[t][k];
      cons[t] = s;
    }
  }

  // ---- M_w = U^H D_w U via reverse conjugation; thread = (w = t>>4, k = t&15) ----
  const int w = t >> 4, k = t & 15;
  for (int c = 0; c < 16; ++c) { Mr[w][k][c] = 0.f; Mi[w][k][c] = 0.f; }
  Mr[w][k][k] = ((k >> (3 - w)) & 1) ? -1.f : 1.f;   // D_w: +1 if qubit-w bit==0
  __syncthreads();

  for (int g = 0; g < 16; ++g) {
    if (g_type[g] == 0) {
      // CNOT conjugation: M <- P M P (P symmetric involution)
      const int pc = 3 - g_a[g], pt = 3 - g_b[g];
      for (int idx = 0; idx < 16; ++idx)         // column swaps within row k
        if (((idx >> pc) & 1) && !((idx >> pt) & 1)) {
          int j = idx | (1 << pt);
          float tr = Mr[w][k][idx]; Mr[w][k][idx] = Mr[w][k][j]; Mr[w][k][j] = tr;
          float ti = Mi[w][k][idx]; Mi[w][k][idx] = Mi[w][k][j]; Mi[w][k][j] = ti;
        }
      __syncthreads();
      for (int idx = 0; idx < 16; ++idx)         // row swaps within column k
        if (((idx >> pc) & 1) && !((idx >> pt) & 1)) {
          int j = idx | (1 << pt);
          float tr = Mr[w][idx][k]; Mr[w][idx][k] = Mr[w][j][k]; Mr[w][j][k] = tr;
          float ti = Mi[w][idx][k]; Mi[w][idx][k] = Mi[w][j][k]; Mi[w][j][k] = ti;
        }
      __syncthreads();
    } else {
      // Rot(phi,theta,omega) on qubit q: g00=(ca ct,-sa ct) g01=(-cb st,-sb st)
      //                                  g10=(cb st,-sb st) g11=(ca ct, sa ct)
      const int l = g_a[g], q = g_b[g];
      float ph = wts[(l*4+q)*3+0], th = wts[(l*4+q)*3+1], om = wts[(l*4+q)*3+2];
      float al = 0.5f*(ph+om), be = 0.5f*(ph-om);
      float ca = cosf(al), sa = sinf(al), cb = cosf(be), sb = sinf(be);
      float ct = cosf(0.5f*th), st = sinf(0.5f*th);
      float g00r= ca*ct, g00i=-sa*ct, g01r=-cb*st, g01i=-sb*st;
      float g10r= cb*st, g10i=-sb*st, g11r= ca*ct, g11i= sa*ct;
      const int p = 3 - q, str = 1 << p;
      // M <- M g  (thread owns row k)
      for (int i = 0; i < 8; ++i) {
        int c0 = ((i >> p) << (p+1)) | (i & (str-1)); int c1 = c0 | str;
        float ar = Mr[w][k][c0], ai = Mi[w][k][c0];
        float br = Mr[w][k][c1], bi = Mi[w][k][c1];
        Mr[w][k][c0] = ar*g00r - ai*g00i + br*g10r - bi*g10i;
        Mi[w][k][c0] = ar*g00i + ai*g00r + br*g10i + bi*g10r;
        Mr[w][k][c1] = ar*g01r - ai*g01i + br*g11r - bi*g11i;
        Mi[w][k][c1] = ar*g01i + ai*g01r + br*g11i + bi*g11r;
      }
      __syncthreads();
      // M <- g^H M  (thread owns column k)
      for (int i = 0; i < 8; ++i) {
        int r0 = ((i >> p) << (p+1)) | (i & (str-1)); int r1 = r0 | str;
        float ar = Mr[w][r0][k], ai = Mi[w][r0][k];
        float br = Mr[w][r1][k], bi = Mi[w][r1][k];
        Mr[w][r0][k] =  g00r*ar + g00i*ai + g10r*br + g10i*bi;
        Mi[w][r0][k] =  g00r*ai - g00i*ar + g10r*bi - g10i*br;
        Mr[w][r1][k] =  g01r*ar + g01i*ai + g11r*br + g11i*bi;
        Mi[w][r1][k] =  g01r*ai - g01i*ar + g11r*bi - g11i*br;
      }
      __syncthreads();
    }
  }

  // ---- Pauli-string coefficients: coeff = Re[Tr(M P_s)]/16 ----
  for (int s = t; s < 4*81; s += 64) {
    int wi = s / 81, ss = s % 81;
    int digs[4] = { ss/27, (ss/9)%3, (ss/3)%3, ss%3 };  // 0=I, 1=Y, 2=Z per qubit
    int ymask = 0, zymask = 0, nY = 0;
    for (int u = 0; u < 4; ++u) {
      int bit = 1 << (3 - u);
      if (digs[u] == 1) { ymask |= bit; zymask |= bit; nY++; }
      else if (digs[u] == 2) zymask |= bit;
    }
    float acc = 0.f;
    for (int r = 0; r < 16; ++r) {
      int c = r ^ ymask;
      float sgn = (__builtin_popcount(r & zymask) & 1) ? -1.f : 1.f;
      acc += sgn * ((nY & 1) ? Mi[wi][r][c] : Mr[wi][r][c]);
    }
    int m4 = nY & 3;                         // Re[i^nY * (Sr + i Si)]
    float f = (m4 == 1 || m4 == 2) ? -1.f : 1.f;
    cons[COEFF_OFF + s] = acc * f * 0.0625f;
  }
}

// ---------------- main kernel ----------------
__global__ void __launch_bounds__(256)
qcircuit_wmma_kernel(const float* __restrict__ x,
                     const float* __restrict__ W,
                     const float* __restrict__ bias,
                     const float* __restrict__ gamma,
                     const float* __restrict__ beta,
                     const float* __restrict__ cons,
                     float* __restrict__ out) {
  const int tid   = blockIdx.x * 256 + threadIdx.x;   // one batch element per lane
  const int lane  = threadIdx.x & 31;
  const int wbase = tid - lane;
  const int row   = lane & 15;
  const bool hi   = lane >= 16;

  // ---- encode: xt = pi*tanh(x); t_u = (1, -sin xt, cos xt) ----
  const v4f xv = __builtin_nontemporal_load((const v4f*)x + tid);  // streamed once (NT)
  const float PI = 3.14159265358979323846f;
  float x0 = PI * f_tanh(xv.x), x1 = PI * f_tanh(xv.y);
  float x2 = PI * f_tanh(xv.z), x3 = PI * f_tanh(xv.w);
  float tS0 = -f_sin(x0), tC0 = f_cos(x0);
  float tS1 = -f_sin(x1), tC1 = f_cos(x1);
  float tS2 = -f_sin(x2), tC2 = f_cos(x2);
  float tS3 = -f_sin(x3), tC3 = f_cos(x3);

  // ---- z_w: multilinear Pauli contraction (80 FMA per output) ----
  float zz[4];
  #pragma unroll
  for (int wq = 0; wq < 4; ++wq) {
    const float* CF = cons + COEFF_OFF + wq*81;
    float A0[3];
    #pragma unroll
    for (int d0 = 0; d0 < 3; ++d0) {
      float A1[3];
      #pragma unroll
      for (int d1 = 0; d1 < 3; ++d1) {
        float A2[3];
        #pragma unroll
        for (int d2 = 0; d2 < 3; ++d2) {
          const float* cc = CF + ((d0*3 + d1)*3 + d2)*3;
          A2[d2] = cc[0] + cc[1]*tS3 + cc[2]*tC3;
        }
        A1[d1] = A2[0] + A2[1]*tS2 + A2[2]*tC2;
      }
      A0[d0] = A1[0] + A1[1]*tS1 + A1[2]*tC1;
    }
    zz[wq] = A0[0] + A0[1]*tS0 + A0[2]*tC0;
  }

  // ---- softmax ----
  float zm = fmaxf(fmaxf(zz[0], zz[1]), fmaxf(zz[2], zz[3]));
  float e0 = f_exp(zz[0]-zm), e1 = f_exp(zz[1]-zm), e2 = f_exp(zz[2]-zm), e3 = f_exp(zz[3]-zm);
  float inv = f_rcp(e0 + e1 + e2 + e3);
  float q0 = e0*inv, q1 = e1*inv, q2 = e2*inv, q3 = e3*inv;

  // ---- closed-form LayerNorm stats from q ----
  float mu = (q0*cons[0] + q1*cons[1] + q2*cons[2] + q3*cons[3] + cons[18]) * 0.015625f;
  float qb  = q0*cons[4] + q1*cons[5] + q2*cons[6] + q3*cons[7];
  float qGq = q0*q0*cons[8] + q1*q1*cons[12] + q2*q2*cons[15] + q3*q3*cons[17]
            + 2.f*(q0*q1*cons[9]  + q0*q2*cons[10] + q0*q3*cons[11]
                 + q1*q2*cons[13] + q1*q3*cons[14] + q2*q3*cons[16]);
  float Eh2 = (qGq + 2.f*qb + cons[19]) * 0.015625f;
  float invstd = f_rsq(Eh2 - mu*mu + 1e-5f);

  // ---- WMMA f32 16x16x4: h-tiles = A(q) * B(W^T); LN fused into NT stores ----
  const int koff = hi ? 2 : 0;
  v2f Bf[4]; float gv[4], bv[4], bb[4];
  #pragma unroll
  for (int j = 0; j < 4; ++j) {
    const int n = j*16 + row;
    float2 wv = *(const float2*)(W + n*4 + koff);   // B frag: lanes<16 K=0,1 ; >=16 K=2,3
    Bf[j] = (v2f){wv.x, wv.y};
    gv[j] = gamma[n]; bv[j] = beta[n]; bb[j] = bias[n];
  }

  #pragma unroll
  for (int a = 0; a < 2; ++a) {
    const int src = row + a*16;                     // A frag: row M=src, K split by lane half
    float t0 = bperm(src, q0), t2 = bperm(src, q2);
    float t1 = bperm(src, q1), t3 = bperm(src, q3);
    v2f Af = (v2f){ hi ? t2 : t0, hi ? t3 : t1 };

    float mur[8], isr[8];                           // per-D-row LN params
    #pragma unroll
    for (int r = 0; r < 8; ++r) {
      const int msrc = a*16 + r + (hi ? 8 : 0);
      mur[r] = bperm(msrc, mu);
      isr[r] = bperm(msrc, invstd);
    }

    float* po = out + (size_t)(wbase + a*16 + (hi ? 8 : 0)) * 64 + row;
    #pragma unroll
    for (int j = 0; j < 4; ++j) {
      v8f Cz = {};
      v8f D = __builtin_amdgcn_wmma_f32_16x16x4_f32(
          false, Af, false, Bf[j], (short)0, Cz, false, false);
      #pragma unroll
      for (int r = 0; r < 8; ++r) {
        float val = (D[r] + bb[j] - mur[r]) * isr[r] * gv[j] + bv[j];
        __builtin_nontemporal_store(val, po + (size_t)r*64 + j*16);  // write-once stream
      }
    }
  }
}

extern "C" void kernel_launch(void* const* d_in, const int* in_sizes, int n_in,
                              void* d_out, int out_size, void* d_ws, size_t ws_size,
                              hipStream_t stream) {
  (void)n_in; (void)out_size; (void)ws_size;
  const float* x     = (const float*)d_in[0];   // (B,4)
  const float* wts   = (const float*)d_in[1];   // (2,4,3)
  const float* W     = (const float*)d_in[2];   // (64,4)
  const float* b     = (const float*)d_in[3];   // (64,)
  const float* gamma = (const float*)d_in[4];   // (64,)
  const float* beta  = (const float*)d_in[5];   // (64,)
  float* out  = (float*)d_out;                  // (B,64)
  float* cons = (float*)d_ws;                   // 356 floats scratch

  qprep<<<1, 64, 0, stream>>>(W, b, wts, cons);
  const int Bn = in_sizes[0] / 4;               // 524288, divisible by 256
  qcircuit_wmma_kernel<<<Bn / 256, 256, 0, stream>>>(x, W, b, gamma, beta, cons, out);
}